// PhiCell_40295383171190
// MI455X (gfx1250) — compile-verified
//
#include <hip/hip_runtime.h>

// ---------------------------------------------------------------------------
// PhiCell hysteresis scan. Phi(y)=max(y,0)+min(y+1,0) implies
//   out_i = min(max(out_{i-1}, x_i), x_i+1) = clamp(out_{i-1}, x_i, x_i+1),
// and clamp transforms compose associatively:
//   (A then B)(v) = clamp(v, clamp(loA,loB,hiB), clamp(hiA,loB,hiB)).
// => parallel prefix scan over (lo,hi) pairs.
// 3-pass: block transforms -> block-start values -> apply + write outputs.
// Bandwidth-bound: ~64MB HBM traffic => ~3us at 23.3 TB/s (pass-3 re-read of
// the 32MB input is L2-resident in the 192MB L2). Tiles staged through LDS
// with gfx1250 async global<->LDS B128 ops (ASYNCcnt + s_wait_asynccnt).
// ---------------------------------------------------------------------------

#define BLOCK 256
#define SEG   32
#define SEGP  36                 // padded per-thread segment stride (floats)
#define TILE  (BLOCK * SEG)      // 8192 elements per block
#define F4PT  (TILE / 4 / BLOCK) // 8 float4 fills per thread
#define P2THREADS 1024

#define INFF __builtin_inff()

#if defined(__has_builtin)
#  if __has_builtin(__builtin_amdgcn_global_load_async_to_lds_b128) && \
      __has_builtin(__builtin_amdgcn_s_wait_asynccnt)
#    define ASYNC_LD 1
#  endif
#  if __has_builtin(__builtin_amdgcn_global_store_async_from_lds_b128) && \
      __has_builtin(__builtin_amdgcn_s_wait_asynccnt)
#    define ASYNC_ST 1
#  endif
#endif
#ifndef ASYNC_LD
#define ASYNC_LD 0
#endif
#ifndef ASYNC_ST
#define ASYNC_ST 0
#endif

// Builtin parameter types (per hipcc diagnostic): v4i in AS1 / AS3.
typedef int v4i __attribute__((vector_size(4 * sizeof(int))));
typedef __attribute__((address_space(1))) v4i* g_v4i_p;
typedef __attribute__((address_space(3))) v4i* l_v4i_p;

__device__ __forceinline__ float clampv(float v, float lo, float hi) {
    return fminf(fmaxf(v, lo), hi);
}

// Element e of the tile lives at lds[(e/32)*SEGP + (e%32)].
// SEGP*4 = 144 bytes and (e%32)%4==0 for f4 chunks => 16B-aligned B128 slots;
// per-thread sequential reads hit banks (t*SEGP + j) % 64 (SEGP=36 => only
// 2-way conflicts), while the fill/writeback is fully coalesced.
__device__ __forceinline__ int lds_index(int e) {
    return ((e >> 5) * SEGP) + (e & 31);
}

// Coalesced tile fill: global -> padded LDS.  Uniform fast path for full
// tiles: 8 back-to-back async B128 loads per thread, no lane guards.
__device__ __forceinline__ void fill_tile(const float* __restrict__ inp,
                                          float* lds, int tileBase, int n,
                                          int tid, bool full) {
    if (full) {
#pragma unroll
        for (int k4 = 0; k4 < F4PT; ++k4) {
            int e = (tid + k4 * BLOCK) << 2;
            int g = tileBase + e;
#if ASYNC_LD
            __builtin_amdgcn_global_load_async_to_lds_b128(
                (g_v4i_p)(v4i*)(inp + g), (l_v4i_p)(v4i*)(lds + lds_index(e)),
                0, 0);
#else
            *(float4*)(lds + lds_index(e)) = *(const float4*)(inp + g);
#endif
        }
    } else {
        for (int k4 = 0; k4 < F4PT; ++k4) {
            int e = (tid + k4 * BLOCK) << 2;
            int g = tileBase + e;
            int lidx = lds_index(e);
#pragma unroll
            for (int j = 0; j < 4; ++j)
                lds[lidx + j] = (g + j) < n ? inp[g + j] : 0.0f;
        }
    }
#if ASYNC_LD
    __builtin_amdgcn_s_wait_asynccnt(0);
#endif
    __syncthreads();
}

// Per-thread compose of SEG element transforms (x, x+1), reading from LDS.
__device__ __forceinline__ void thread_transform(const float* lds, int base,
                                                 int gseg, int n, float k,
                                                 bool full, float& lo,
                                                 float& hi) {
    lo = -INFF; hi = INFF;
    if (full) {
#pragma unroll 8
        for (int j = 0; j < SEG; ++j) {
            float x = lds[base + j] * k;
            lo = clampv(lo, x, x + 1.0f);
            hi = clampv(hi, x, x + 1.0f);
        }
    } else {
        for (int j = 0; j < SEG; ++j) {
            if (gseg + j < n) {
                float x = lds[base + j] * k;
                lo = clampv(lo, x, x + 1.0f);
                hi = clampv(hi, x, x + 1.0f);
            }
        }
    }
}

// Ordered inclusive Hillis-Steele scan of clamp transforms over BLOCK threads.
__device__ __forceinline__ void block_scan(float* slo, float* shi, int tid,
                                           float lo, float hi) {
    slo[tid] = lo; shi[tid] = hi;
    __syncthreads();
    for (int s = 1; s < BLOCK; s <<= 1) {
        float aLo = 0.f, aHi = 0.f, bLo = 0.f, bHi = 0.f;
        if (tid >= s) { aLo = slo[tid - s]; aHi = shi[tid - s];
                        bLo = slo[tid];     bHi = shi[tid]; }
        __syncthreads();
        if (tid >= s) {  // compose: earlier (a) then current (b)
            slo[tid] = clampv(aLo, bLo, bHi);
            shi[tid] = clampv(aHi, bLo, bHi);
        }
        __syncthreads();
    }
}

// Pass 1: per-block composed transform -> ws
__global__ void __launch_bounds__(BLOCK)
phicell_pass1(const float* __restrict__ inp, const float* __restrict__ kptr,
              float2* __restrict__ tf, int n) {
    __shared__ float lds[BLOCK * SEGP];
    __shared__ float slo[BLOCK], shi[BLOCK];
    int tid = threadIdx.x;
    int tileBase = blockIdx.x * TILE;
    bool full = (tileBase + TILE) <= n;
    float k = kptr[0];
    fill_tile(inp, lds, tileBase, n, tid, full);

    float lo, hi;
    thread_transform(lds, tid * SEGP, tileBase + tid * SEG, n, k, full, lo, hi);
    block_scan(slo, shi, tid, lo, hi);
    if (tid == BLOCK - 1)
        tf[blockIdx.x] = make_float2(slo[BLOCK - 1], shi[BLOCK - 1]);
}

// Pass 2: scan block transforms -> per-block starting values + final state
__global__ void __launch_bounds__(P2THREADS)
phicell_pass2(const float2* __restrict__ tf, float* __restrict__ starts,
              const float* __restrict__ statePtr, float* __restrict__ stateOut,
              int nblk) {
    __shared__ float slo[P2THREADS], shi[P2THREADS];
    int tid = threadIdx.x;
    float carry = statePtr[0];  // running state value (uniform across threads)
    for (int basei = 0; basei < nblk; basei += P2THREADS) {
        int b = basei + tid;
        float lo = -INFF, hi = INFF;
        if (b < nblk) { float2 t = tf[b]; lo = t.x; hi = t.y; }
        slo[tid] = lo; shi[tid] = hi;
        __syncthreads();
        for (int s = 1; s < P2THREADS; s <<= 1) {
            float aLo = 0.f, aHi = 0.f, bLo = 0.f, bHi = 0.f;
            if (tid >= s) { aLo = slo[tid - s]; aHi = shi[tid - s];
                            bLo = slo[tid];     bHi = shi[tid]; }
            __syncthreads();
            if (tid >= s) {
                slo[tid] = clampv(aLo, bLo, bHi);
                shi[tid] = clampv(aHi, bLo, bHi);
            }
            __syncthreads();
        }
        float exLo = -INFF, exHi = INFF;   // exclusive prefix for this block
        if (tid > 0) { exLo = slo[tid - 1]; exHi = shi[tid - 1]; }
        if (b < nblk) starts[b] = clampv(carry, exLo, exHi);
        int rem = nblk - basei;
        int last = (rem < P2THREADS ? rem : P2THREADS) - 1;
        float tLo = slo[last], tHi = shi[last];
        __syncthreads();
        carry = clampv(carry, tLo, tHi);
    }
    if (tid == 0) stateOut[0] = carry;
}

// Pass 3: recompute thread transforms, apply, write outputs
__global__ void __launch_bounds__(BLOCK)
phicell_pass3(const float* __restrict__ inp, const float* __restrict__ kptr,
              const float* __restrict__ starts, float* __restrict__ outp, int n) {
    __shared__ float lds[BLOCK * SEGP];
    __shared__ float slo[BLOCK], shi[BLOCK];
    int tid = threadIdx.x;
    int tileBase = blockIdx.x * TILE;
    bool full = (tileBase + TILE) <= n;
    float k = kptr[0];
    fill_tile(inp, lds, tileBase, n, tid, full);

    float lo, hi;
    int base = tid * SEGP;
    int gseg = tileBase + tid * SEG;
    thread_transform(lds, base, gseg, n, k, full, lo, hi);
    block_scan(slo, shi, tid, lo, hi);

    float vb = starts[blockIdx.x];
    float exLo = -INFF, exHi = INFF;
    if (tid > 0) { exLo = slo[tid - 1]; exHi = shi[tid - 1]; }
    float v = clampv(vb, exLo, exHi);   // state entering this thread's segment

    if (full) {                          // cheap value recurrence, in-place
#pragma unroll 8
        for (int j = 0; j < SEG; ++j) {
            float x = lds[base + j] * k;
            v = clampv(v, x, x + 1.0f);
            lds[base + j] = v;
        }
    } else {
        for (int j = 0; j < SEG; ++j) {
            if (gseg + j < n) {
                float x = lds[base + j] * k;
                v = clampv(v, x, x + 1.0f);
                lds[base + j] = v;
            }
        }
    }
    __syncthreads();

    // Coalesced writeback: padded LDS -> global (async-from-LDS fast path).
    if (full) {
#pragma unroll
        for (int k4 = 0; k4 < F4PT; ++k4) {
            int e = (tid + k4 * BLOCK) << 2;
            int g = tileBase + e;
#if ASYNC_ST
            __builtin_amdgcn_global_store_async_from_lds_b128(
                (g_v4i_p)(v4i*)(outp + g), (l_v4i_p)(v4i*)(lds + lds_index(e)),
                0, 0);
#else
            *(float4*)(outp + g) = *(const float4*)(lds + lds_index(e));
#endif
        }
    } else {
        for (int k4 = 0; k4 < F4PT; ++k4) {
            int e = (tid + k4 * BLOCK) << 2;
            int g = tileBase + e;
            int lidx = lds_index(e);
            for (int j = 0; j < 4; ++j)
                if (g + j < n) outp[g + j] = lds[lidx + j];
        }
    }
#if ASYNC_ST
    __builtin_amdgcn_s_wait_asynccnt(0);
#endif
}

extern "C" void kernel_launch(void* const* d_in, const int* in_sizes, int n_in,
                              void* d_out, int out_size, void* d_ws, size_t ws_size,
                              hipStream_t stream) {
    const float* inp      = (const float*)d_in[0];  // inputs [1,T]
    const float* statePtr = (const float*)d_in[1];  // state  [1,1]
    const float* kptr     = (const float*)d_in[2];  // kernel [1,1]
    float* outp = (float*)d_out;                    // [T outputs][1 state]

    int n = in_sizes[0];
    int nblk = (n + TILE - 1) / TILE;               // 1024 for T=2^23

    float2* tf     = (float2*)d_ws;                                 // nblk pairs
    float*  starts = (float*)((char*)d_ws + (size_t)nblk * sizeof(float2));

    phicell_pass1<<<nblk, BLOCK, 0, stream>>>(inp, kptr, tf, n);
    phicell_pass2<<<1, P2THREADS, 0, stream>>>(tf, starts, statePtr, outp + n, nblk);
    phicell_pass3<<<nblk, BLOCK, 0, stream>>>(inp, kptr, starts, outp, n);
}